// Structure_Decoder_17824114279157
// MI455X (gfx1250) — compile-verified
//
#include <hip/hip_runtime.h>
#include <hip/hip_bf16.h>

typedef __attribute__((ext_vector_type(2))) float v2f;
typedef __attribute__((ext_vector_type(8))) float v8f;

#define NHID 128
#define LDSS 132   // LDS row stride in floats (128 -> 132: A/B fragment loads conflict-free)

// ---------------------------------------------------------------------------
// Kernel 1: hw = h @ W   (64 rows of h per block, full 128 output cols)
// 8 waves: wave w computes N-subtile w (16 cols) for 4 M-subtiles.
// ---------------------------------------------------------------------------
__global__ __launch_bounds__(256) void hw_gemm_kernel(const float* __restrict__ h,
                                                      const float* __restrict__ W,
                                                      float* __restrict__ hw, int n) {
    __shared__ float sH[64 * LDSS];
    __shared__ float sW[128 * LDSS];
    const int bm  = blockIdx.x * 64;
    const int tid = threadIdx.x;

    for (int i = tid; i < 64 * 32; i += 256) {
        int row = i >> 5, c4 = i & 31;
        int gr = bm + row; if (gr >= n) gr = n - 1;           // clamp; stores are bounded
        float4 v = ((const float4*)(h + (size_t)gr * NHID))[c4];
        *(float4*)&sH[row * LDSS + c4 * 4] = v;
    }
    for (int i = tid; i < 128 * 32; i += 256) {
        int row = i >> 5, c4 = i & 31;
        float4 v = ((const float4*)(W + (size_t)row * NHID))[c4];
        *(float4*)&sW[row * LDSS + c4 * 4] = v;
    }
    __syncthreads();

    const int wave = tid >> 5, lane = tid & 31;
    const int half = lane >> 4, lm = lane & 15;
    const int ncol = wave * 16;

    v8f acc[4] = {};
    const float* pb = &sW[ncol + lm];                         // B[k][n] = W[k][n]
    for (int k0 = 0; k0 < NHID; k0 += 4) {
        const int kk = k0 + 2 * half;
        v2f b; b.x = pb[(size_t)kk * LDSS]; b.y = pb[(size_t)(kk + 1) * LDSS];
#pragma unroll
        for (int m = 0; m < 4; ++m) {
            const float* pa = &sH[(m * 16 + lm) * LDSS + kk];
            v2f a; a.x = pa[0]; a.y = pa[1];
            acc[m] = __builtin_amdgcn_wmma_f32_16x16x4_f32(false, a, false, b,
                                                           (short)0, acc[m], false, false);
        }
    }

    const int col = ncol + lm;
    if (bm + 64 <= n) {
#pragma unroll
        for (int m = 0; m < 4; ++m)
#pragma unroll
            for (int i = 0; i < 8; ++i)
                hw[(size_t)(bm + m * 16 + i + 8 * half) * NHID + col] = acc[m][i];
    } else {
#pragma unroll
        for (int m = 0; m < 4; ++m)
#pragma unroll
            for (int i = 0; i < 8; ++i) {
                int row = bm + m * 16 + i + 8 * half;
                if (row < n) hw[(size_t)row * NHID + col] = acc[m][i];
            }
    }
}

// ---------------------------------------------------------------------------
// Kernel 2a: zero agg (ws is poisoned by the harness)
// ---------------------------------------------------------------------------
__global__ __launch_bounds__(256) void zero_kernel(float4* __restrict__ p, int n4) {
    int t = blockIdx.x * 256 + threadIdx.x;
    if (t < n4) p[t] = make_float4(0.f, 0.f, 0.f, 0.f);
}

// ---------------------------------------------------------------------------
// Kernel 2b: agg[dst] += hw[src]   (one thread per edge x 4-channel chunk)
// ---------------------------------------------------------------------------
__global__ __launch_bounds__(256) void scatter_kernel(const float* __restrict__ hw,
                                                      const int* __restrict__ esrc,
                                                      const int* __restrict__ edst,
                                                      float* __restrict__ agg, int n_edges) {
    long long t = (long long)blockIdx.x * 256 + threadIdx.x;
    int e = (int)(t >> 5);
    int c = ((int)t & 31) * 4;
    if (e >= n_edges) return;
    int s = esrc[e], d = edst[e];
    float4 v = *(const float4*)(hw + (size_t)s * NHID + c);
    float* p = agg + (size_t)d * NHID + c;
    unsafeAtomicAdd(p + 0, v.x);
    unsafeAtomicAdd(p + 1, v.y);
    unsafeAtomicAdd(p + 2, v.z);
    unsafeAtomicAdd(p + 3, v.w);
}

// ---------------------------------------------------------------------------
// Kernel 3: x = relu(agg + b), in place
// ---------------------------------------------------------------------------
__global__ __launch_bounds__(256) void bias_relu_kernel(float* __restrict__ agg,
                                                        const float* __restrict__ bias,
                                                        int total) {
    int t = blockIdx.x * 256 + threadIdx.x;
    if (t < total) {
        float v = agg[t] + bias[t & (NHID - 1)];
        agg[t] = v > 0.f ? v : 0.f;
    }
}

// ---------------------------------------------------------------------------
// Kernel 4: out = x @ x.T   (128x128 output tile / block, WMMA f32 16x16x4)
// 8 waves as 4(M) x 2(N): each wave owns 32 rows x 64 cols = 8 C tiles.
// Per k-step: 2 A-frags + 4 B-frags -> 8 WMMA.
// ---------------------------------------------------------------------------
__global__ __launch_bounds__(256) void xxt_kernel(const float* __restrict__ x,
                                                  float* __restrict__ out, int n) {
    __shared__ float sA[128 * LDSS];
    __shared__ float sB[128 * LDSS];
    const int bm  = blockIdx.x * 128;
    const int bn  = blockIdx.y * 128;
    const int tid = threadIdx.x;

    // Stage 128x128 A rows and B rows (B = x rows for the column tile).
    for (int i = tid; i < 128 * 32; i += 256) {
        int row = i >> 5, c4 = i & 31;
        int ga = bm + row; if (ga >= n) ga = n - 1;
        int gb = bn + row; if (gb >= n) gb = n - 1;
        float4 va = ((const float4*)(x + (size_t)ga * NHID))[c4];
        float4 vb = ((const float4*)(x + (size_t)gb * NHID))[c4];
        *(float4*)&sA[row * LDSS + c4 * 4] = va;
        *(float4*)&sB[row * LDSS + c4 * 4] = vb;
    }
    __syncthreads();

    const int wave = tid >> 5, lane = tid & 31;
    const int wm = wave >> 1;          // 0..3 -> 32 output rows each
    const int wn = wave & 1;           // 0..1 -> 64 output cols each
    const int half = lane >> 4, lm = lane & 15;

    v8f acc[2][4] = {};
    const float* pa0 = &sA[(wm * 32 +      lm) * LDSS];
    const float* pa1 = &sA[(wm * 32 + 16 + lm) * LDSS];
    const float* pb0 = &sB[(wn * 64 +      lm) * LDSS];   // B[k][j] = x[j][k]
    const float* pb1 = &sB[(wn * 64 + 16 + lm) * LDSS];
    const float* pb2 = &sB[(wn * 64 + 32 + lm) * LDSS];
    const float* pb3 = &sB[(wn * 64 + 48 + lm) * LDSS];

    for (int k0 = 0; k0 < NHID; k0 += 4) {
        const int kk = k0 + 2 * half;
        v2f a0; a0.x = pa0[kk]; a0.y = pa0[kk + 1];
        v2f a1; a1.x = pa1[kk]; a1.y = pa1[kk + 1];
        v2f b0; b0.x = pb0[kk]; b0.y = pb0[kk + 1];
        v2f b1; b1.x = pb1[kk]; b1.y = pb1[kk + 1];
        v2f b2; b2.x = pb2[kk]; b2.y = pb2[kk + 1];
        v2f b3; b3.x = pb3[kk]; b3.y = pb3[kk + 1];
        acc[0][0] = __builtin_amdgcn_wmma_f32_16x16x4_f32(false, a0, false, b0, (short)0, acc[0][0], false, false);
        acc[0][1] = __builtin_amdgcn_wmma_f32_16x16x4_f32(false, a0, false, b1, (short)0, acc[0][1], false, false);
        acc[0][2] = __builtin_amdgcn_wmma_f32_16x16x4_f32(false, a0, false, b2, (short)0, acc[0][2], false, false);
        acc[0][3] = __builtin_amdgcn_wmma_f32_16x16x4_f32(false, a0, false, b3, (short)0, acc[0][3], false, false);
        acc[1][0] = __builtin_amdgcn_wmma_f32_16x16x4_f32(false, a1, false, b0, (short)0, acc[1][0], false, false);
        acc[1][1] = __builtin_amdgcn_wmma_f32_16x16x4_f32(false, a1, false, b1, (short)0, acc[1][1], false, false);
        acc[1][2] = __builtin_amdgcn_wmma_f32_16x16x4_f32(false, a1, false, b2, (short)0, acc[1][2], false, false);
        acc[1][3] = __builtin_amdgcn_wmma_f32_16x16x4_f32(false, a1, false, b3, (short)0, acc[1][3], false, false);
    }

    const int colb = bn + wn * 64 + lm;
    const int rowb = bm + wm * 32;
    if (bm + 128 <= n && bn + 128 <= n) {
        // Interior fast path: unconditional stores, no exec-mask churn.
#pragma unroll
        for (int mi = 0; mi < 2; ++mi)
#pragma unroll
            for (int nj = 0; nj < 4; ++nj) {
                float* po = out + (size_t)(rowb + mi * 16 + 8 * half) * n + colb + nj * 16;
#pragma unroll
                for (int i = 0; i < 8; ++i)
                    po[(size_t)i * n] = acc[mi][nj][i];
            }
    } else {
#pragma unroll
        for (int mi = 0; mi < 2; ++mi)
#pragma unroll
            for (int nj = 0; nj < 4; ++nj) {
                int col = colb + nj * 16;
                if (col >= n) continue;
#pragma unroll
                for (int i = 0; i < 8; ++i) {
                    int row = rowb + mi * 16 + i + 8 * half;
                    if (row < n) out[(size_t)row * n + col] = acc[mi][nj][i];
                }
            }
    }
}

// ---------------------------------------------------------------------------
extern "C" void kernel_launch(void* const* d_in, const int* in_sizes, int n_in,
                              void* d_out, int out_size, void* d_ws, size_t ws_size,
                              hipStream_t stream) {
    const float* h    = (const float*)d_in[0];
    const float* W    = (const float*)d_in[1];
    const float* bias = (const float*)d_in[2];
    const int* esrc   = (const int*)d_in[3];
    const int* edst   = (const int*)d_in[4];
    float* out = (float*)d_out;

    const int n       = in_sizes[0] / NHID;   // 10000
    const int n_edges = in_sizes[3];          // 320000
    const int total   = n * NHID;

    float* hw  = (float*)d_ws;                // [n,128]
    float* agg = hw + (size_t)total;          // [n,128]

    // 1) hw = h @ W
    hw_gemm_kernel<<<(n + 63) / 64, 256, 0, stream>>>(h, W, hw, n);

    // 2) agg = segment_sum(hw[src] -> dst)
    zero_kernel<<<(total / 4 + 255) / 256, 256, 0, stream>>>((float4*)agg, total / 4);
    long long sth = (long long)n_edges * 32;
    scatter_kernel<<<(unsigned)((sth + 255) / 256), 256, 0, stream>>>(hw, esrc, edst, agg, n_edges);

    // 3) x = relu(agg + b), in place
    bias_relu_kernel<<<(total + 255) / 256, 256, 0, stream>>>(agg, bias, total);

    // 4) out = x @ x.T
    dim3 g((n + 127) / 128, (n + 127) / 128);
    xxt_kernel<<<g, 256, 0, stream>>>(agg, out, n);
}